// WindowMajorityModel_46995532153210
// MI455X (gfx1250) — compile-verified
//
#include <hip/hip_runtime.h>

// ---------------------------------------------------------------------------
// WindowMajorityModel for MI455X (gfx1250).
//
// Cost model: output is 8*2048*32000 f32 = ~2.1 GB of stores -> ~90us at
// 23.3 TB/s. Histogram/argmax over 16K tokens is noise. So:
//   Phase 1: tiny per-row LDS histogram + packed-key argmax (8 blocks).
//   Phase 2: bandwidth fill. Build the per-row 32000-float pattern (125 KB,
//            fits gfx1250's 320 KB WGP LDS) once per block, then stream it to
//            global memory with the CDNA5 async LDS->global store engine
//            (global_store_async_from_lds_b128, tracked by ASYNCcnt):
//            contiguous 512 B per wave-instruction, NT hint so the 2.1 GB
//            stream doesn't thrash the 192 MB L2.
//
// aaa_async_probe is a tiny never-launched kernel placed FIRST in the TU so
// the disasm snippet conclusively shows whether the async-store path lowered
// (builtin branch) or which asm syntax works (inline-asm branch).
// ---------------------------------------------------------------------------

#define VOCAB 32000
#define BSZ   8
#define SEQ   2048
#define BOS   1

typedef __attribute__((ext_vector_type(4))) int   v4i;
typedef __attribute__((ext_vector_type(4))) float v4f;

// ---- Probe: shows the async LDS->global store path in the snippet ---------
extern "C" __global__ void __launch_bounds__(64) aaa_async_probe(v4i* out) {
#if defined(__gfx1250__)
  __shared__ v4i lbuf[64];
  const int t = threadIdx.x;
  lbuf[t] = v4i{0, 0, 0, 0};
  __syncthreads();
#if __has_builtin(__builtin_amdgcn_global_store_async_from_lds_b128)
  typedef __attribute__((address_space(1))) v4i* gp1;
  typedef __attribute__((address_space(3))) v4i* lp3;
  __builtin_amdgcn_global_store_async_from_lds_b128((gp1)out + t, (lp3)lbuf + t, 0, 1);
#else
  unsigned long long ga = (unsigned long long)out + (unsigned long long)t * 16ull;
  unsigned la = (unsigned)(unsigned long long)&lbuf[t];   // low 32 bits = LDS offset
  asm volatile("global_store_async_from_lds_b128 %0, %1, off"
               :: "v"(ga), "v"(la) : "memory");
#endif
#if __has_builtin(__builtin_amdgcn_s_wait_asynccnt)
  __builtin_amdgcn_s_wait_asynccnt(0);
#else
  asm volatile("s_wait_asynccnt 0" ::: "memory");
#endif
#endif
}

// ---- Phase 2: bandwidth fill ----------------------------------------------
#define FILL_THREADS 1024
#define SCHUNK 8   // sequence positions written per block

__global__ __launch_bounds__(FILL_THREADS) void fill_kernel(const int* __restrict__ pred,
                                                            float* __restrict__ out) {
  extern __shared__ float pat[];               // VOCAB floats = 128000 B of LDS
  const int tid = threadIdx.x;
  const int b   = blockIdx.z;
  const int s0  = blockIdx.y * SCHUNK;
  const int p   = pred[b];

  for (int v = tid; v < VOCAB; v += FILL_THREADS)
    pat[v] = (v == p) ? 6.0f : -6.0f;
  __syncthreads();                             // ds writes visible before DMA reads

#if defined(__gfx1250__) && __has_builtin(__builtin_amdgcn_global_store_async_from_lds_b128)
  typedef __attribute__((address_space(1))) v4i* gp1;
  typedef __attribute__((address_space(3))) v4i* lp3;
  gp1 gout = (gp1)out;
  lp3 lpat = (lp3)pat;
  for (int s = 0; s < SCHUNK; ++s) {
    const size_t base = ((size_t)b * SEQ + (size_t)(s0 + s)) * (VOCAB / 4);
    for (int i = tid; i < VOCAB / 4; i += FILL_THREADS) {
      // 16 B per lane, lanes contiguous -> 512 B per wave instruction,
      // tracked on ASYNCcnt. cpol = 1 (TH_NT): non-temporal streaming store.
      __builtin_amdgcn_global_store_async_from_lds_b128(gout + base + i, lpat + i, 0, 1);
    }
  }
#if __has_builtin(__builtin_amdgcn_s_wait_asynccnt)
  __builtin_amdgcn_s_wait_asynccnt(0);
#else
  asm volatile("s_wait_asynccnt 0" ::: "memory");
#endif
#else
  // Fallback (host pass / toolchains without the async builtin):
  // plain 128-bit vector stores of the LDS pattern.
  v4f* o = (v4f*)out;
  const v4f* lp = (const v4f*)pat;
  for (int s = 0; s < SCHUNK; ++s) {
    const size_t base = ((size_t)b * SEQ + (size_t)(s0 + s)) * (VOCAB / 4);
    for (int i = tid; i < VOCAB / 4; i += FILL_THREADS)
      o[base + i] = lp[i];
  }
#endif
}

// ---- Phase 1: per-row histogram + argmax ----------------------------------
// Packed key = (count << 15) | (32767 - v):
//   max over keys => max count; ties => larger inverted index => lower v,
//   matching jnp.argmax tie-to-lowest. count <= 2048 so key < 2^27.
__global__ __launch_bounds__(1024) void hist_kernel(const int* __restrict__ ids,
                                                    int* __restrict__ pred) {
  extern __shared__ int cnt[];                 // VOCAB ints = 128000 B of LDS
  const int tid = threadIdx.x;
  const int b   = blockIdx.x;

  for (int v = tid; v < VOCAB; v += 1024) cnt[v] = 0;
  __syncthreads();

  const int* row = ids + (size_t)b * SEQ;
  for (int i = tid; i < SEQ; i += 1024) {
    int t = row[i];
    if (t >= 2) atomicAdd(&cnt[t], 1);         // exclude pad(0) and BOS(1)
  }
  __syncthreads();

  int best = 0;
  for (int v = tid; v < VOCAB; v += 1024) {
    int key = (cnt[v] << 15) | (32767 - v);
    best = max(best, key);
  }
  __syncthreads();
  // cnt[0] holds count of token 0, which is always 0 -> safe reduction cell.
  atomicMax(&cnt[0], best);
  __syncthreads();
  if (tid == 0) {
    int key = cnt[0];
    pred[b] = (key >> 15) == 0 ? BOS : (32767 - (key & 0x7fff));
  }
}

// ---------------------------------------------------------------------------
extern "C" void kernel_launch(void* const* d_in, const int* in_sizes, int n_in,
                              void* d_out, int out_size, void* d_ws, size_t ws_size,
                              hipStream_t stream) {
  (void)in_sizes; (void)n_in; (void)out_size; (void)ws_size;
  const int* ids = (const int*)d_in[0];        // [8, 2048] int32
  float*     out = (float*)d_out;              // [8, 2048, 32000] f32
  int*       pred = (int*)d_ws;                // 8 ints of scratch

  hipLaunchKernelGGL(hist_kernel, dim3(BSZ), dim3(1024),
                     VOCAB * sizeof(int), stream, ids, pred);

  dim3 grid(1, SEQ / SCHUNK, BSZ);             // (1, 256, 8) blocks
  hipLaunchKernelGGL(fill_kernel, grid, dim3(FILL_THREADS),
                     VOCAB * sizeof(float), stream, pred, out);
}